// TopKRegression_85048942395529
// MI455X (gfx1250) — compile-verified
//
#include <hip/hip_runtime.h>
#include <math.h>

// ---------------------------------------------------------------------------
// TopK(k=2)-softmax disparity regression, MI455X (gfx1250).
// Memory-bound: 402.7 MB read + 8.4 MB write @ 23.3 TB/s ~= 17.6 us floor.
// Strategy: TDM (tensor_load_to_lds) DMAs one 48x256 f32 tile (48 KB) per
// workgroup into LDS asynchronously (TENSORcnt); 4 wave32 waves then scan
// LDS conflict-free (float2 per lane -> ds_load_b64, banks 2t/2t+1) for
// top-2 + softmax expectation. Cross-block occupancy (~6 WGs/WGP by LDS)
// overlaps DMA of some blocks with compute of others.
// ---------------------------------------------------------------------------

typedef unsigned int v4u __attribute__((ext_vector_type(4)));
typedef int          v4i __attribute__((ext_vector_type(4)));
typedef int          v8i __attribute__((ext_vector_type(8)));

#ifndef __has_builtin
#define __has_builtin(x) 0
#endif

#if __has_builtin(__builtin_amdgcn_tensor_load_to_lds)
#define HAVE_TDM 1
#else
#define HAVE_TDM 0
#endif

#define DD   48         // disparity channels
#define PIX  256        // pixels per tile
#define NT   128        // threads per block (4 waves); 2 pixels per thread
#define HW   131072     // H*W = 256*512

__device__ __forceinline__ void top2_step(float v, int d,
                                          float& v0, float& v1, int& i0, int& i1)
{
    bool gt0 = v > v0;     // strict '>' => earliest index wins ties (lax.top_k)
    bool gt1 = v > v1;
    v1 = gt0 ? v0 : (gt1 ? v : v1);
    i1 = gt0 ? i0 : (gt1 ? d : i1);
    v0 = gt0 ? v  : v0;
    i0 = gt0 ? d  : i0;
}

__device__ __forceinline__ float disp_from_top2(float v0, float v1, int i0, int i1)
{
    // softmax over {v0,v1}, v1<=v0: e = exp(v1-v0) in (0,1], no overflow
    float e = __expf(v1 - v0);
    return fmaf((float)i1, e, (float)i0) / (1.0f + e);
}

extern "C" __global__ __launch_bounds__(NT)
void topk2_disp_kernel(const float* __restrict__ cost, float* __restrict__ out)
{
    extern __shared__ float smem[];               // DD*PIX floats, dynamic LDS => offset 0
    const int t = threadIdx.x;
    const long long P = (long long)blockIdx.x * PIX;   // tile's first pixel in [0, B*HW)
    const int b = (int)(P / HW);
    const int p = (int)(P % HW);                       // PIX | HW => tile never straddles b

    // tile start: cost[b][0][p .. p+PIX)
    const float* tile = cost + ((long long)b * DD * HW + p);

#if HAVE_TDM
    if (t < 32) {                                  // wave 0 only: one DMA per block
        unsigned long long gaddr = (unsigned long long)(const void*)tile;

        // ---- D# group 0 (ISA 8.3): count=1, lds_addr=0, global_addr, type=2
        v4u g0;
        g0.x = 1u;                                           // count=1 (valid), user mode
        g0.y = 0u;                                           // lds_addr = 0
        g0.z = (unsigned int)(gaddr & 0xFFFFFFFFu);          // global_addr[31:0]
        g0.w = (unsigned int)((gaddr >> 32) & 0x01FFFFFFu)   // global_addr[56:32]
             | (2u << 30);                                   // type = 2 ("image")

        // ---- D# group 1 (ISA 8.4): data_size=4B, dims/strides, tile dims
        v8i g1;
        g1[0] = (int)(2u << 16);                             // data_size=2 (4 bytes)
        g1[1] = (int)((unsigned)PIX << 16);                  // tensor_dim0[15:0]
        g1[2] = (int)((((unsigned)PIX >> 16) & 0xFFFFu)      // tensor_dim0[31:16]
                      | ((unsigned)DD << 16));               // tensor_dim1[15:0]
        g1[3] = (int)((((unsigned)DD >> 16) & 0xFFFFu)       // tensor_dim1[31:16]
                      | ((unsigned)PIX << 16));              // tile_dim0 = 256
        g1[4] = (int)((unsigned)DD);                         // tile_dim1=48, tile_dim2=0
        g1[5] = (int)((unsigned)HW);                         // tensor_dim0_stride[31:0]
        g1[6] = 0;                                           // stride0[47:32] | stride1 lo
        g1[7] = 0;                                           // stride1 hi
        v4i g2 = {0, 0, 0, 0};                               // 2D tensor: groups 2/3 unused
        v4i g3 = {0, 0, 0, 0};

#if __clang_major__ >= 23
        v8i g4 = {0, 0, 0, 0, 0, 0, 0, 0};
        __builtin_amdgcn_tensor_load_to_lds(g0, g1, g2, g3, g4, 0);
#else
        __builtin_amdgcn_tensor_load_to_lds(g0, g1, g2, g3, 0);
#endif
        __builtin_amdgcn_s_wait_tensorcnt(0);      // issuing wave waits for DMA done
    }
    __syncthreads();                               // publish LDS tile to all waves
#else
    // Fallback: cooperative coalesced global->LDS staging (2 floats per step).
    {
        float2* s2w = (float2*)smem;
        for (int i = t; i < DD * (PIX / 2); i += NT) {
            int d = i / (PIX / 2), x = (i - d * (PIX / 2)) * 2;
            s2w[i] = *(const float2*)(tile + (long long)d * HW + x);
        }
    }
    __syncthreads();
#endif

    // ---- top-2 over d; 2 pixels/lane => ds_load_b64, two independent chains
    const float2* s2 = (const float2*)smem;        // PIX/2 float2 per d-row
    float a0 = -__builtin_huge_valf(), a1 = -__builtin_huge_valf();
    float c0 = -__builtin_huge_valf(), c1 = -__builtin_huge_valf();
    int   ai0 = 0, ai1 = 0, ci0 = 0, ci1 = 0;
#pragma unroll
    for (int d = 0; d < DD; ++d) {
        float2 v = s2[d * (PIX / 2) + t];          // lane t -> banks 2t,2t+1: conflict-free
        top2_step(v.x, d, a0, a1, ai0, ai1);
        top2_step(v.y, d, c0, c1, ci0, ci1);
    }

    float2 r;
    r.x = disp_from_top2(a0, a1, ai0, ai1);
    r.y = disp_from_top2(c0, c1, ci0, ci1);
    ((float2*)(out + P))[t] = r;                   // global_store_b64, 8B aligned
}

extern "C" void kernel_launch(void* const* d_in, const int* in_sizes, int n_in,
                              void* d_out, int out_size, void* d_ws, size_t ws_size,
                              hipStream_t stream)
{
    const float* cost = (const float*)d_in[0];     // [16,48,256,512] f32
    // d_in[1] is k==2 (baked in)
    float* out = (float*)d_out;                    // [16,1,256,512] f32

    const int nblocks = out_size / PIX;            // 2,097,152 / 256 = 8192
    hipLaunchKernelGGL(topk2_disp_kernel, dim3(nblocks), dim3(NT),
                       DD * PIX * sizeof(float), stream, cost, out);
}